// WaveletTransform3D_44547400794821
// MI455X (gfx1250) — compile-verified
//
#include <hip/hip_runtime.h>
#include <cstdint>
#include <cstddef>

// ---------------------------------------------------------------------------
// 3D Haar wavelet (frames: valid, H/W: 'same' right-zero-pad), fully fused.
// x: (2,3,16,256,256) f32  ->  LLL (2,3,15,256,256) ++ detail (2,3,105,256,256)
// Memory-bound (~214 MB @ 23.3 TB/s): optimize data movement, not FLOPs.
// CDNA5 path: global_load_async_to_lds_b128 staging + s_wait_asynccnt,
// non-temporal b128 stores for the 189 MB write stream.
// ---------------------------------------------------------------------------

typedef float v4f __attribute__((ext_vector_type(4)));

#define BCN    6          // batch*channel
#define F_IN   16
#define F_OUT  15
#define HH_    256
#define WW_    256
#define FRAME  (HH_ * WW_)
#define TH     16         // output rows per tile
#define NTILE  (HH_ / TH) // 16
#define W4     (WW_ / 4)  // 64 float4 per row
#define SC     0.35355339059327373f   // 1/(2*sqrt(2)) per stage (matches ref rounding)

__global__ __launch_bounds__(256)
void haar3d_fused_kernel(const float* __restrict__ x, float* __restrict__ out)
{
    // Tile: 2 frames x 17 rows x 256 cols of f32 = 34816 B LDS
    __shared__ __align__(16) float lds[2 * 17 * WW_];

    const int tid = threadIdx.x;
    int blk = blockIdx.x;
    const int ht = blk % NTILE;  blk /= NTILE;
    const int f  = blk % F_OUT;  blk /= F_OUT;
    const int bc = blk;                    // 0..5
    const int h0 = ht * TH;

    // LDS byte offset of tile base (low 32 bits of the flat LDS address)
    const unsigned ldsBase = (unsigned)(uintptr_t)(&lds[0]);
    // Uniform 64-bit global base for this (b,c): goes into an SGPR pair
    const uint64_t srcBase = (uint64_t)(x + (size_t)bc * (F_IN * FRAME));

    // ---- Stage 2 frames x 17 rows into LDS with async B128 copies ----------
    // 2*17*64 = 2176 float4 chunks, striped over 256 threads.
    for (int i = tid; i < 2 * 17 * W4; i += 256) {
        const int fr  = i / (17 * W4);
        const int rem = i - fr * (17 * W4);
        const int row = rem / W4;
        const int c4  = rem - row * W4;
        const int h   = h0 + row;
        const unsigned loff = ldsBase + (unsigned)(((fr * 17 + row) * WW_ + c4 * 4) * 4);
        if (h < HH_) {
            const unsigned goff =
                (unsigned)((((f + fr) * HH_ + h) * WW_ + c4 * 4) * sizeof(float));
            asm volatile("global_load_async_to_lds_b128 %0, %1, %2"
                         :: "v"(loff), "v"(goff), "s"(srcBase)
                         : "memory");
        } else {
            // bottom 'same' pad row (only last tile): zero-fill in LDS
            *(v4f*)&lds[(fr * 17 + row) * WW_ + c4 * 4] = (v4f)0.0f;
        }
    }
    asm volatile("s_wait_asynccnt 0" ::: "memory");
    __syncthreads();

    // ---- Output bases -------------------------------------------------------
    float* outLLL = out + ((size_t)(bc * F_OUT + f)) * FRAME;
    // detail tensor starts after LLL; layout (bc, s*15+f, h, w), s = 0..6
    float* outD = out + (size_t)BCN * F_OUT * FRAME
                      + ((size_t)bc * 7 * F_OUT + f) * FRAME;
    const size_t dstr = (size_t)F_OUT * FRAME;   // stride between subbands

    // ---- Butterflies: 16 rows x 64 quad-cols, striped over 256 threads -----
    for (int i = tid; i < TH * W4; i += 256) {
        const int r  = i / W4;
        const int c4 = i - r * W4;
        const int w  = c4 * 4;
        const int h  = h0 + r;

        const float* L0 = &lds[(0 * 17 + r    ) * WW_];  // frame f,   row h
        const float* L1 = &lds[(0 * 17 + r + 1) * WW_];  // frame f,   row h+1
        const float* M0 = &lds[(1 * 17 + r    ) * WW_];  // frame f+1, row h
        const float* M1 = &lds[(1 * 17 + r + 1) * WW_];  // frame f+1, row h+1

        v4f a0 = *(const v4f*)(L0 + w);
        v4f a1 = *(const v4f*)(L1 + w);
        v4f b0 = *(const v4f*)(M0 + w);
        v4f b1 = *(const v4f*)(M1 + w);
        const bool edge = (c4 == W4 - 1);            // w+4 == 256 -> zero pad
        float a0n = edge ? 0.0f : L0[w + 4];
        float a1n = edge ? 0.0f : L1[w + 4];
        float b0n = edge ? 0.0f : M0[w + 4];
        float b1n = edge ? 0.0f : M1[w + 4];

        // stage 1: frames (valid):  L = S*(A+B), H = S*(B-A)
        v4f s0 = SC * (a0 + b0), d0 = SC * (b0 - a0);
        v4f s1 = SC * (a1 + b1), d1 = SC * (b1 - a1);
        float sn0 = SC * (a0n + b0n), dn0 = SC * (b0n - a0n);
        float sn1 = SC * (a1n + b1n), dn1 = SC * (b1n - a1n);

        // stage 2: height ('same'): low = S*(row_h + row_h1), high = S*(row_h1 - row_h)
        v4f t = SC * (s0 + s1), u = SC * (s1 - s0);
        v4f v = SC * (d0 + d1), z = SC * (d1 - d0);
        float tn = SC * (sn0 + sn1), un = SC * (sn1 - sn0);
        float vn = SC * (dn0 + dn1), zn = SC * (dn1 - dn0);

        // stage 3: width ('same') with one-element shift
        v4f ts = {t.y, t.z, t.w, tn};
        v4f us = {u.y, u.z, u.w, un};
        v4f vs = {v.y, v.z, v.w, vn};
        v4f zs = {z.y, z.z, z.w, zn};

        v4f LLL = SC * (t + ts), LLH = SC * (ts - t);
        v4f LHL = SC * (u + us), LHH = SC * (us - u);
        v4f HLL = SC * (v + vs), HLH = SC * (vs - v);
        v4f HHL = SC * (z + zs), HHH = SC * (zs - z);

        const size_t pix = (size_t)h * WW_ + w;
        __builtin_nontemporal_store(LLL, (v4f*)(outLLL + pix));
        float* db = outD + pix;
        __builtin_nontemporal_store(LLH, (v4f*)(db + 0 * dstr));
        __builtin_nontemporal_store(LHL, (v4f*)(db + 1 * dstr));
        __builtin_nontemporal_store(LHH, (v4f*)(db + 2 * dstr));
        __builtin_nontemporal_store(HLL, (v4f*)(db + 3 * dstr));
        __builtin_nontemporal_store(HLH, (v4f*)(db + 4 * dstr));
        __builtin_nontemporal_store(HHL, (v4f*)(db + 5 * dstr));
        __builtin_nontemporal_store(HHH, (v4f*)(db + 6 * dstr));
    }
}

extern "C" void kernel_launch(void* const* d_in, const int* in_sizes, int n_in,
                              void* d_out, int out_size, void* d_ws, size_t ws_size,
                              hipStream_t stream) {
    const float* x = (const float*)d_in[0];
    float* out = (float*)d_out;
    (void)in_sizes; (void)n_in; (void)out_size; (void)d_ws; (void)ws_size;

    const int nblocks = BCN * F_OUT * NTILE;   // 6*15*16 = 1440
    haar3d_fused_kernel<<<nblocks, 256, 0, stream>>>(x, out);
}